// PopulationGNN_31593779429377
// MI455X (gfx1250) — compile-verified
//
#include <hip/hip_runtime.h>
#include <hip/hip_bf16.h>

typedef float v2f __attribute__((ext_vector_type(2)));
typedef float v8f __attribute__((ext_vector_type(8)));

#define BN_EPS    1e-5f
#define RES_SCALE 0.7f
#define TILE_M    32
#define KC        64

// ---------------------------------------------------------------------------
// Generic zero kernel
// ---------------------------------------------------------------------------
__global__ void zero_f32(float* __restrict__ p, long long n) {
    long long t = (long long)blockIdx.x * blockDim.x + threadIdx.x;
    if (t < n) p[t] = 0.0f;
}

// ---------------------------------------------------------------------------
// Degree accumulation: deg[dst[e]] += 1
// ---------------------------------------------------------------------------
__global__ void deg_accum(const long long* __restrict__ dst, float* __restrict__ deg,
                          long long E) {
    long long t = (long long)blockIdx.x * blockDim.x + threadIdx.x;
    if (t < E) atomicAdd(&deg[dst[t]], 1.0f);
}

// dinv = rsqrt(deg + 1)   (in-place on deg buffer)
__global__ void deg_to_dinv(float* __restrict__ deg, int n) {
    int t = blockIdx.x * blockDim.x + threadIdx.x;
    if (t < n) deg[t] = rsqrtf(deg[t] + 1.0f);
}

// ---------------------------------------------------------------------------
// WMMA f32 GEMM:  out[nrows x 64] = X[nrows x K] @ W[K x 64]
// Block: 256 threads = 8 waves; each wave owns a 16x16 output tile.
//   waves: colTile = wave&3 (4 x 16 = 64 cols), rowTile = wave>>2 (2 x 16 rows)
// A and B chunks staged through LDS. The full-chunk path (kc == 64) is fully
// unrolled: 16 back-to-back v_wmma_f32_16x16x4_f32 with batched ds_loads so
// LDS latency overlaps matrix-pipe issue instead of serializing on
// s_wait_dscnt 0 per WMMA.
// nrows must be a multiple of 16 (50000 = 3125*16). K padded per-chunk.
// ---------------------------------------------------------------------------
__global__ __launch_bounds__(256) void gemm_wmma_f32(
    const float* __restrict__ X, const float* __restrict__ W,
    float* __restrict__ out, int nrows, int K)
{
    __shared__ float As[TILE_M * KC];   //  8 KB
    __shared__ float Bs[KC * 64];       // 16 KB

    const int tid     = threadIdx.x;
    const int lane    = tid & 31;
    const int wave    = tid >> 5;
    const int colTile = wave & 3;
    const int rowTile = wave >> 2;
    const int rowBase = blockIdx.x * TILE_M;
    const int waveRow = rowBase + rowTile * 16;
    const int m       = lane & 15;   // row (A) / col (B) within fragment
    const int khalf   = lane >> 4;   // selects K pair {0,1} or {2,3}

    v8f c = {};

    for (int k0 = 0; k0 < K; k0 += KC) {
        const int kc = (K - k0 < KC) ? (K - k0) : KC;   // 64, or 16 remainder

        // Stage B chunk: W[k0 .. k0+kc-1][0..63], zero-pad tail
        for (int i = tid; i < KC * 64; i += 256) {
            int r = i >> 6, cc = i & 63;
            Bs[i] = (r < kc) ? W[(size_t)(k0 + r) * 64 + cc] : 0.0f;
        }
        // Stage A chunk: X[rowBase .. rowBase+31][k0 .. k0+kc-1], zero-pad
        for (int i = tid; i < TILE_M * KC; i += 256) {
            int r = i >> 6, cc = i & 63;
            int gr = rowBase + r;
            As[i] = (gr < nrows && cc < kc) ? X[(size_t)gr * K + k0 + cc] : 0.0f;
        }
        __syncthreads();

        if (waveRow < nrows) {   // wave-uniform: EXEC stays all-ones for WMMA
            const int ar = rowTile * 16 + m;
            const int bn = colTile * 16 + m;
            if (kc == KC) {
                // Full chunk: fully unrolled, 16 WMMAs, pipelined LDS loads.
                #pragma unroll
                for (int kk = 0; kk < KC; kk += 4) {
                    v2f a, b;
                    a.x = As[ar * KC + kk + 2 * khalf + 0];
                    a.y = As[ar * KC + kk + 2 * khalf + 1];
                    b.x = Bs[(kk + 2 * khalf + 0) * 64 + bn];
                    b.y = Bs[(kk + 2 * khalf + 1) * 64 + bn];
                    c = __builtin_amdgcn_wmma_f32_16x16x4_f32(
                            false, a, false, b, (short)0, c, false, false);
                }
            } else {
                // Remainder chunk (kc = 16 for K=2000): unroll by 4.
                #pragma unroll 4
                for (int kk = 0; kk < kc; kk += 4) {
                    v2f a, b;
                    a.x = As[ar * KC + kk + 2 * khalf + 0];
                    a.y = As[ar * KC + kk + 2 * khalf + 1];
                    b.x = Bs[(kk + 2 * khalf + 0) * 64 + bn];
                    b.y = Bs[(kk + 2 * khalf + 1) * 64 + bn];
                    c = __builtin_amdgcn_wmma_f32_16x16x4_f32(
                            false, a, false, b, (short)0, c, false, false);
                }
            }
        }
        __syncthreads();
    }

    if (waveRow < nrows) {
        // C/D layout: VGPR r -> M = r + 8*(lane>=16), N = lane&15
        const int outRow = waveRow + (lane >> 4) * 8;
        const int col    = colTile * 16 + (lane & 15);
        #pragma unroll
        for (int r = 0; r < 8; ++r)
            out[(size_t)(outRow + r) * 64 + col] = c[r];
    }
}

// ---------------------------------------------------------------------------
// Self-loop + bias init: agg[i][f] = xw[i][f] * dinv[i]^2 + bias[f]
// ---------------------------------------------------------------------------
__global__ void agg_init(const float* __restrict__ xw, const float* __restrict__ dinv,
                         const float* __restrict__ bias, float* __restrict__ agg,
                         long long total) {
    long long t = (long long)blockIdx.x * blockDim.x + threadIdx.x;
    if (t >= total) return;
    long long i = t >> 6;
    int f = (int)(t & 63);
    float di = dinv[i];
    agg[t] = xw[t] * di * di + bias[f];
}

// ---------------------------------------------------------------------------
// Edge scatter: one wave per edge, 2 features per lane.
//   agg[dst] += xw[src] * dinv[src]*dinv[dst]
// ---------------------------------------------------------------------------
__global__ __launch_bounds__(256) void edge_agg(
    const long long* __restrict__ src, const long long* __restrict__ dst,
    const float* __restrict__ dinv, const float* __restrict__ xw,
    float* __restrict__ agg, long long E)
{
    long long t = (long long)blockIdx.x * blockDim.x + threadIdx.x;
    long long e = t >> 5;
    if (e >= E) return;
    int lane = threadIdx.x & 31;
    long long s = src[e], d = dst[e];
    float w = dinv[s] * dinv[d];
    float2 v = ((const float2*)(xw + s * 64))[lane];
    atomicAdd(&agg[d * 64 + lane * 2 + 0], v.x * w);
    atomicAdd(&agg[d * 64 + lane * 2 + 1], v.y * w);
}

// ---------------------------------------------------------------------------
// BN statistics: per-column sum and sum-of-squares (64 columns)
// ---------------------------------------------------------------------------
__global__ __launch_bounds__(256) void bn_stats(
    const float* __restrict__ x, float* __restrict__ sums,
    float* __restrict__ sumsq, int n)
{
    __shared__ float ls[256];
    __shared__ float lq[256];
    const int col  = threadIdx.x & 63;
    const int rsub = threadIdx.x >> 6;   // 0..3
    float s = 0.0f, q = 0.0f;
    for (int r = blockIdx.x * 4 + rsub; r < n; r += gridDim.x * 4) {
        float v = x[(size_t)r * 64 + col];
        s += v; q += v * v;
    }
    ls[threadIdx.x] = s; lq[threadIdx.x] = q;
    __syncthreads();
    if (threadIdx.x < 64) {
        s = ls[threadIdx.x] + ls[threadIdx.x + 64] + ls[threadIdx.x + 128] + ls[threadIdx.x + 192];
        q = lq[threadIdx.x] + lq[threadIdx.x + 64] + lq[threadIdx.x + 128] + lq[threadIdx.x + 192];
        atomicAdd(&sums[threadIdx.x], s);
        atomicAdd(&sumsq[threadIdx.x], q);
    }
}

// ---------------------------------------------------------------------------
// Softmax of L layer-mixing weights (tiny; single thread)
// ---------------------------------------------------------------------------
__global__ void softmax_small(const float* __restrict__ lw, float* __restrict__ wsm, int L) {
    if (threadIdx.x == 0 && blockIdx.x == 0) {
        float mx = lw[0];
        for (int i = 1; i < L; ++i) mx = fmaxf(mx, lw[i]);
        float sum = 0.0f;
        for (int i = 0; i < L; ++i) sum += __expf(lw[i] - mx);
        float inv = 1.0f / sum;
        for (int i = 0; i < L; ++i) wsm[i] = __expf(lw[i] - mx) * inv;
    }
}

// ---------------------------------------------------------------------------
// Fused BN-normalize + ReLU + residual + weighted emb accumulation
// ---------------------------------------------------------------------------
__global__ void bn_norm(const float* __restrict__ agg, const float* __restrict__ prev,
                        const float* __restrict__ gamma, const float* __restrict__ beta,
                        const float* __restrict__ sums, const float* __restrict__ sumsq,
                        const float* __restrict__ wsm,
                        float* __restrict__ cur, float* __restrict__ emb,
                        int n, int layer)
{
    long long t = (long long)blockIdx.x * blockDim.x + threadIdx.x;
    if (t >= (long long)n * 64) return;
    int col = (int)(t & 63);
    float invn = 1.0f / (float)n;
    float m = sums[col] * invn;
    float v = sumsq[col] * invn - m * m;
    float y = (agg[t] - m) * rsqrtf(v + BN_EPS) * gamma[col] + beta[col];
    y = fmaxf(y, 0.0f);
    if (layer > 0) y += RES_SCALE * prev[t];
    cur[t] = y;
    float wv = wsm[layer] * y;
    if (layer == 0) emb[t] = wv;
    else            emb[t] += wv;
}

// ---------------------------------------------------------------------------
// Host orchestration
// ---------------------------------------------------------------------------
extern "C" void kernel_launch(void* const* d_in, const int* in_sizes, int n_in,
                              void* d_out, int out_size, void* d_ws, size_t ws_size,
                              hipStream_t stream) {
    const float*      features = (const float*)d_in[0];
    const long long*  edges    = (const long long*)d_in[1];
    const float*      W0       = (const float*)d_in[2];
    const float*      b0       = (const float*)d_in[3];
    const float*      Wh       = (const float*)d_in[4];
    const float*      bh       = (const float*)d_in[5];
    const float*      gamma    = (const float*)d_in[6];
    const float*      beta     = (const float*)d_in[7];
    const float*      lw       = (const float*)d_in[8];

    const int       HID    = in_sizes[3];              // 64
    const int       IN_DIM = in_sizes[2] / HID;        // 2000
    const int       N      = in_sizes[0] / IN_DIM;     // 50000
    const long long E      = in_sizes[1] / 2;          // 1.6e6
    const int       L      = in_sizes[6] / HID;        // 4

    const long long* esrc = edges;
    const long long* edst = edges + E;

    // Workspace layout (floats)
    float* f     = (float*)d_ws;
    float* dinv  = f;                         // N
    float* xw    = dinv + N;                  // N*64
    float* agg   = xw + (long long)N * 64;    // N*64
    float* bufA  = agg + (long long)N * 64;   // N*64
    float* bufB  = bufA + (long long)N * 64;  // N*64
    float* sums  = bufB + (long long)N * 64;  // 64
    float* sumsq = sums + 64;                 // 64
    float* wsm   = sumsq + 64;                // L

    float* emb = (float*)d_out;

    const long long NF = (long long)N * 64;
    const int blkElem  = (int)((NF + 255) / 256);

    // --- degrees / dinv ---
    zero_f32<<<(N + 255) / 256, 256, 0, stream>>>(dinv, N);
    deg_accum<<<(unsigned)((E + 255) / 256), 256, 0, stream>>>(edst, dinv, E);
    deg_to_dinv<<<(N + 255) / 256, 256, 0, stream>>>(dinv, N);

    // --- softmax of layer-mixing weights ---
    softmax_small<<<1, 32, 0, stream>>>(lw, wsm, L);

    float* prev = nullptr;
    float* cur  = bufA;

    for (int layer = 0; layer < L; ++layer) {
        const float* X  = (layer == 0) ? features : prev;
        const float* W  = (layer == 0) ? W0 : (Wh + (size_t)(layer - 1) * HID * HID);
        const float* bb = (layer == 0) ? b0 : (bh + (size_t)(layer - 1) * HID);
        const int    K  = (layer == 0) ? IN_DIM : HID;

        // xw = X @ W (WMMA f32)
        gemm_wmma_f32<<<(N + TILE_M - 1) / TILE_M, 256, 0, stream>>>(X, W, xw, N, K);

        // agg = xw * snorm + bias, then scatter edges
        agg_init<<<blkElem, 256, 0, stream>>>(xw, dinv, bb, agg, NF);
        {
            long long tthreads = E * 32;
            edge_agg<<<(unsigned)((tthreads + 255) / 256), 256, 0, stream>>>(
                esrc, edst, dinv, xw, agg, E);
        }

        // BN stats
        zero_f32<<<1, 128, 0, stream>>>(sums, 128);   // sums + sumsq contiguous
        bn_stats<<<256, 256, 0, stream>>>(agg, sums, sumsq, N);

        // Normalize + ReLU + residual + emb accumulation
        bn_norm<<<blkElem, 256, 0, stream>>>(
            agg, prev, gamma + (size_t)layer * HID, beta + (size_t)layer * HID,
            sums, sumsq, wsm, cur, emb, N, layer);

        // ping-pong layer buffers
        prev = cur;
        cur  = (cur == bufA) ? bufB : bufA;
    }
}